// NeRFRenderer_52149492908742
// MI455X (gfx1250) — compile-verified
//
#include <hip/hip_runtime.h>
#include <hip/hip_bf16.h>

typedef __attribute__((ext_vector_type(2))) float v2f;
typedef __attribute__((ext_vector_type(8))) float v8f;

static constexpr int kP = 128;  // samples per ray
static constexpr int kF = 64;   // feature dim

// One wave (32 lanes) handles one ray end-to-end.
__global__ __launch_bounds__(32) void nerf_render_kernel(
    const float* __restrict__ sigma,   // [HW, P]
    const float* __restrict__ feat,    // [HW, P, F]
    const float* __restrict__ depth,   // [HW, P]
    const float* __restrict__ rays,    // [HW, 3]
    float* __restrict__ feat_map,      // [F, HW]
    float* __restrict__ depth_map,     // [HW]
    float* __restrict__ alpha_map,     // [HW]
    float* __restrict__ weights_out,   // [HW, P]
    int HW)
{
  __shared__ unsigned long long s[kP];  // (depthBits<<32)|origIdx sort keys
  __shared__ float sig[kP];             // sigma in original order
  __shared__ float wu[kP];              // weights scattered back to original order
  __shared__ float wsort[kP];           // weights in sorted order

  const int r = blockIdx.x;
  const int t = threadIdx.x;            // 0..31
  const int base = r * kP;

  // ---- load depth + sigma (coalesced), pack keys for stable argsort ----
#pragma unroll
  for (int k = 0; k < 4; ++k) {
    int i = t + 32 * k;
    float d = depth[base + i];
    sig[i] = sigma[base + i];
    s[i] = ((unsigned long long)__float_as_uint(d) << 32) | (unsigned)i;
  }
  __syncthreads();

  // ---- bitonic sort of 128 u64 keys (depth>0 so uint order == float order) ----
  for (int k = 2; k <= kP; k <<= 1) {
    for (int j = k >> 1; j > 0; j >>= 1) {
#pragma unroll
      for (int q = 0; q < 4; ++q) {
        int i = t + 32 * q;
        int ixj = i ^ j;
        if (ixj > i) {
          unsigned long long a = s[i], b = s[ixj];
          bool up = ((i & k) == 0);
          if ((a > b) == up) { s[i] = b; s[ixj] = a; }
        }
      }
      __syncthreads();
    }
  }

  // ---- blocked phase: lane t owns sorted samples [4t, 4t+4) ----
  const int i0 = 4 * t;
  unsigned long long s0 = s[i0], s1 = s[i0 + 1], s2 = s[i0 + 2], s3 = s[i0 + 3];
  float d0 = __uint_as_float((unsigned)(s0 >> 32));
  float d1 = __uint_as_float((unsigned)(s1 >> 32));
  float d2 = __uint_as_float((unsigned)(s2 >> 32));
  float d3 = __uint_as_float((unsigned)(s3 >> 32));
  float dnext = (t < 31) ? __uint_as_float((unsigned)(s[i0 + 4] >> 32)) : 0.0f;

  float rx = rays[3 * r + 0], ry = rays[3 * r + 1], rz = rays[3 * r + 2];
  float rn = sqrtf(rx * rx + ry * ry + rz * rz);

  float dist0 = (d1 - d0) * rn;
  float dist1 = (d2 - d1) * rn;
  float dist2 = (d3 - d2) * rn;
  float dist3 = ((t < 31) ? (dnext - d3) : 1e10f) * rn;

  float g0 = sig[(unsigned)s0 & (kP - 1)];
  float g1 = sig[(unsigned)s1 & (kP - 1)];
  float g2 = sig[(unsigned)s2 & (kP - 1)];
  float g3 = sig[(unsigned)s3 & (kP - 1)];

  float e0 = expf(-fmaxf(g0, 0.f) * dist0);
  float e1 = expf(-fmaxf(g1, 0.f) * dist1);
  float e2 = expf(-fmaxf(g2, 0.f) * dist2);
  float e3 = expf(-fmaxf(g3, 0.f) * dist3);
  float a0 = 1.f - e0, a1 = 1.f - e1, a2 = 1.f - e2, a3 = 1.f - e3;
  float m0 = e0 + 1e-10f, m1 = e1 + 1e-10f, m2 = e2 + 1e-10f, m3 = e3 + 1e-10f;

  // exclusive product scan across the wave (transmittance)
  float lp = m0 * m1 * m2 * m3;
  float incl = lp;
#pragma unroll
  for (int off = 1; off < 32; off <<= 1) {
    float u = __shfl_up(incl, off, 32);
    if (t >= off) incl *= u;
  }
  float excl = __shfl_up(incl, 1, 32);
  if (t == 0) excl = 1.f;

  float tr0 = excl, tr1 = tr0 * m0, tr2 = tr1 * m1, tr3 = tr2 * m2;
  float w0 = a0 * tr0, w1 = a1 * tr1, w2 = a2 * tr2, w3 = a3 * tr3;

  wsort[i0 + 0] = w0; wsort[i0 + 1] = w1; wsort[i0 + 2] = w2; wsort[i0 + 3] = w3;
  wu[(unsigned)s0 & (kP - 1)] = w0;
  wu[(unsigned)s1 & (kP - 1)] = w1;
  wu[(unsigned)s2 & (kP - 1)] = w2;
  wu[(unsigned)s3 & (kP - 1)] = w3;

  float sw   = w0 + w1 + w2 + w3;
  float sabs = fabsf(w0) + fabsf(w1) + fabsf(w2) + fabsf(w3);
  float swd  = w0 * d0 + w1 * d1 + w2 * d2 + w3 * d3;
#pragma unroll
  for (int off = 16; off > 0; off >>= 1) {
    sw   += __shfl_xor(sw, off, 32);
    sabs += __shfl_xor(sabs, off, 32);
    swd  += __shfl_xor(swd, off, 32);
  }
  if (t == 0) {
    alpha_map[r] = sw;
    float dm = swd / fmaxf(sabs, 1e-12f);
    dm = (dm - 0.001f) / (1000.0f - 0.001f);
    depth_map[r] = fminf(fmaxf(dm, 0.f), 1.f);
  }
  __syncthreads();

  // coalesced weights store (sorted order, matches reference)
#pragma unroll
  for (int k = 0; k < 4; ++k) {
    int i = t + 32 * k;
    weights_out[base + i] = wsort[i];
  }

  // ---- feature reduction via WMMA f32 16x16x4:
  // out[f] = sum_p wu[p] * feat[r][p][f], features on the M axis,
  // weight chunk replicated across N. feat streamed in natural order.
  const float* fr = feat + (size_t)base * kF;
  const int g = t >> 4;        // half-wave: K pair selector
  const int l16 = t & 15;      // row within A / col within D
  v8f acc0 = {}, acc1 = {}, acc2 = {}, acc3 = {};
  for (int c = 0; c < 32; ++c) {
    int p0 = c << 2;
    v2f b;                                   // B[k][n] = wu[p0+k], all n
    b.x = wu[p0 + 2 * g];
    b.y = wu[p0 + 2 * g + 1];
    const float* rowA = fr + ((p0 + 2 * g) << 6);  // feat row p0+2g
    const float* rowB = rowA + kF;                 // feat row p0+2g+1
    v2f a0, a1, a2, a3;                            // A[m][k] = feat[p][fb+m]
    a0.x = rowA[l16];      a0.y = rowB[l16];
    a1.x = rowA[l16 + 16]; a1.y = rowB[l16 + 16];
    a2.x = rowA[l16 + 32]; a2.y = rowB[l16 + 32];
    a3.x = rowA[l16 + 48]; a3.y = rowB[l16 + 48];
    acc0 = __builtin_amdgcn_wmma_f32_16x16x4_f32(false, a0, false, b, (short)0, acc0, false, false);
    acc1 = __builtin_amdgcn_wmma_f32_16x16x4_f32(false, a1, false, b, (short)0, acc1, false, false);
    acc2 = __builtin_amdgcn_wmma_f32_16x16x4_f32(false, a2, false, b, (short)0, acc2, false, false);
    acc3 = __builtin_amdgcn_wmma_f32_16x16x4_f32(false, a3, false, b, (short)0, acc3, false, false);
  }
  // D: lane L, VGPR v holds out[fb + v + 8*(L>=16)], identical over n = L&15
  if (l16 == 0) {
    int fo = 8 * g;
#pragma unroll
    for (int v = 0; v < 8; ++v) {
      feat_map[(size_t)( 0 + fo + v) * HW + r] = acc0[v];
      feat_map[(size_t)(16 + fo + v) * HW + r] = acc1[v];
      feat_map[(size_t)(32 + fo + v) * HW + r] = acc2[v];
      feat_map[(size_t)(48 + fo + v) * HW + r] = acc3[v];
    }
  }
}

extern "C" void kernel_launch(void* const* d_in, const int* in_sizes, int n_in,
                              void* d_out, int out_size, void* d_ws, size_t ws_size,
                              hipStream_t stream) {
  const float* sigma = (const float*)d_in[0];
  const float* feat  = (const float*)d_in[1];
  const float* depth = (const float*)d_in[2];
  // d_in[3] = semantic: sorted by the reference but unused in any output -> skipped
  const float* rays  = (const float*)d_in[4];

  const int HW = in_sizes[0] / kP;  // B*H*W rays (B=1)

  float* out       = (float*)d_out;
  float* feat_map  = out;                          // [F, HW]
  float* depth_map = out + (size_t)kF * HW;        // [HW]
  float* alpha_map = depth_map + HW;               // [HW]
  float* weights   = alpha_map + HW;               // [HW, P]

  nerf_render_kernel<<<HW, 32, 0, stream>>>(sigma, feat, depth, rays,
                                            feat_map, depth_map, alpha_map,
                                            weights, HW);
}